// DispLoss_15144054686558
// MI455X (gfx1250) — compile-verified
//
#include <hip/hip_runtime.h>
#include <cstdint>
#include <cstddef>

namespace {
constexpr int kB      = 2;
constexpr int kH      = 384;
constexpr int kW      = 1216;
constexpr int kBins   = 256;
constexpr int kHW     = kH * kW;            // 466944 (multiple of 4)
constexpr int kN      = kB * kHW;           // 933888
constexpr int kGroups = kN / 4;             // 233472 4-pixel groups
constexpr int kBlock  = 256;                // 8 wave32s per workgroup
constexpr int kGrid   = (kGroups + kBlock - 1) / kBlock;  // 912
constexpr int kChunk  = 8;                  // bins staged per async batch
constexpr int kChunks = kBins / kChunk;     // 32 batches
constexpr int kPix    = kBlock * 4;         // 1024 pixels per workgroup
}

__device__ __forceinline__ float waveReduceAdd(float v) {
  #pragma unroll
  for (int off = 16; off > 0; off >>= 1) v += __shfl_down(v, off, 32);
  return v;
}

// Fixed-order block reduction of a triple; true on thread 0 with block sums
// in (a,b,c). Deterministic across graph replays (no float atomics anywhere).
__device__ __forceinline__ bool blockReduce3(float& a, float& b, float& c) {
  __shared__ float red[3][kBlock / 32];
  a = waveReduceAdd(a);
  b = waveReduceAdd(b);
  c = waveReduceAdd(c);
  const int lane = threadIdx.x & 31;
  const int wv   = threadIdx.x >> 5;
  if (lane == 0) { red[0][wv] = a; red[1][wv] = b; red[2][wv] = c; }
  __syncthreads();
  if (threadIdx.x != 0) return false;
  float ra = 0.f, rb = 0.f, rc = 0.f;
  #pragma unroll
  for (int i = 0; i < kBlock / 32; ++i) {
    ra += red[0][i]; rb += red[1][i]; rc += red[2][i];
  }
  a = ra; b = rb; c = rc;
  return true;
}

__global__ __launch_bounds__(kBlock) void disp_loss_main(
    const float*   __restrict__ coord,
    const float*   __restrict__ logits,
    const float*   __restrict__ disp,
    const uint8_t* __restrict__ valid,
    float*         __restrict__ partial)  // [kGrid][3]
{
  // Staging tile: [buffer][bin-in-chunk][pixel]. Each wave owns a disjoint
  // 128-pixel slice (tid-indexed), so only per-wave ASYNCcnt waits are
  // needed -- no cross-wave barriers in the pipeline.
  __shared__ float tile[2][kChunk][kPix];   // 64 KB of 320 KB LDS

  float cntAcc = 0.f, coordAcc = 0.f, ceAcc = 0.f;

  const int g = blockIdx.x * kBlock + threadIdx.x;
  if (g < kGroups) {
    const int p0  = g * 4;            // 4 consecutive pixels, same batch & row
    const int b   = p0 / kHW;
    const int hw0 = p0 - b * kHW;
    const int w0  = hw0 % kW;

    const float4 co = *reinterpret_cast<const float4*>(coord + p0);
    const float4 dp = *reinterpret_cast<const float4*>(disp + p0);
    const uchar4 vl = *reinterpret_cast<const uchar4*>(valid + p0);
    const float  cj[4] = {co.x, co.y, co.z, co.w};
    const float  dj[4] = {dp.x, dp.y, dp.z, dp.w};
    const uint8_t vj[4] = {vl.x, vl.y, vl.z, vl.w};

    // ---- per-pixel mask, L1 term, bin indices / weights (hoisted) ----
    const float invInterval = 255.0f / (1.1f * (float)kW);
    float maskf[4], wh[4];
    int   lb[4], hb[4];
    #pragma unroll
    for (int j = 0; j < 4; ++j) {
      const float target = (float)(w0 + j) - dj[j];
      const float mf = (vj[j] != 0 && dj[j] < 192.0f) ? 1.0f : 0.0f;
      maskf[j] = mf;
      cntAcc   += mf;
      coordAcc += fabsf(cj[j] - target) * mf;

      const float labels = fminf(fmaxf(target + 0.1f * (float)kW, 0.0f),
                                 1.1f * (float)kW);
      const float pos = labels * invInterval;        // in [0, 255]
      const float fl  = floorf(pos);
      int l = (int)fl;
      l = max(0, min(l, kBins - 1));
      lb[j] = l;
      hb[j] = min(l + 1, kBins - 1);
      wh[j] = pos - fl;
    }

    // ---- async-staged single-pass online log-sum-exp over 256 bins ----
    const size_t   base  = (size_t)b * kBins * kHW + (size_t)hw0;
    const uint32_t base4 = (uint32_t)(base * 4u);          // byte offset (<2GB)
    const uint32_t binB  = (uint32_t)(kHW * 4);            // bytes per bin step
    const uint32_t tid4  = (uint32_t)threadIdx.x * 4u;

    // Issue one chunk of kChunk bins x 4 pixels/lane into buffer `buf`.
    auto issueChunk = [&](int c0, int buf) {
      // Close WAR window: recycled buffer's ds_loads must be done before the
      // async engine may overwrite it (DScnt and ASYNCcnt are unordered).
      asm volatile("s_wait_dscnt 0x0" ::: "memory");
      #pragma unroll
      for (int cc = 0; cc < kChunk; ++cc) {
        const uint32_t ldsA =
            (uint32_t)(uintptr_t)&tile[buf][cc][tid4];     // LDS byte address
        const uint32_t gOff = base4 + (uint32_t)(c0 + cc) * binB;
        asm volatile(
            "global_load_async_to_lds_b128 %0, %1, %2"     // ASYNCcnt-tracked
            :: "v"(ldsA), "v"(gOff), "s"(logits)
            : "memory");
      }
    };

    float m[4], s[4];
    #pragma unroll
    for (int j = 0; j < 4; ++j) { m[j] = -__builtin_inff(); s[j] = 0.f; }

    issueChunk(0, 0);                                      // prologue
    for (int ck = 0; ck < kChunks; ++ck) {
      const int buf = ck & 1;
      if (ck + 1 < kChunks) {
        issueChunk((ck + 1) * kChunk, buf ^ 1);            // prefetch next
        // Async loads complete in order: <=kChunk outstanding means the
        // current buffer's kChunk loads have all landed in LDS.
        asm volatile("s_wait_asynccnt 0x8" ::: "memory");
      } else {
        asm volatile("s_wait_asynccnt 0x0" ::: "memory");
      }
      #pragma unroll
      for (int cc = 0; cc < kChunk; ++cc) {
        const float4 x4 =
            *reinterpret_cast<const float4*>(&tile[buf][cc][tid4]); // ds_load_b128
        const float xv[4] = {x4.x, x4.y, x4.z, x4.w};
        #pragma unroll
        for (int j = 0; j < 4; ++j) {
          const float d = xv[j] - m[j];                    // first iter: +inf
          const float e = __expf(-fabsf(d));               // 1 trans op/elt
          const float sGrew = __builtin_fmaf(s[j], e, 1.0f); // new-max case
          const float sSame = s[j] + e;                      // max unchanged
          s[j] = (d > 0.f) ? sGrew : sSame;
          m[j] = fmaxf(m[j], xv[j]);
        }
      }
    }

    // ---- CE taps (cachelines just streamed -> L0/L2 hits) + accumulate ----
    #pragma unroll
    for (int j = 0; j < 4; ++j) {
      const float lse = m[j] + __logf(s[j]);
      const float xl  = logits[base + (size_t)lb[j] * kHW + j];
      const float xh  = logits[base + (size_t)hb[j] * kHW + j];
      const float ce  = lse - ((1.0f - wh[j]) * xl + wh[j] * xh);
      ceAcc += ce * maskf[j];
    }
  }

  if (blockReduce3(cntAcc, coordAcc, ceAcc)) {
    partial[blockIdx.x * 3 + 0] = cntAcc;
    partial[blockIdx.x * 3 + 1] = coordAcc;
    partial[blockIdx.x * 3 + 2] = ceAcc;
  }
}

__global__ __launch_bounds__(kBlock) void disp_loss_finalize(
    const float* __restrict__ partial, float* __restrict__ out)
{
  float a = 0.f, b = 0.f, c = 0.f;
  for (int i = threadIdx.x; i < kGrid; i += kBlock) {
    a += partial[3 * i + 0];
    b += partial[3 * i + 1];
    c += partial[3 * i + 2];
  }
  if (blockReduce3(a, b, c)) {
    const float msum       = a + 1e-6f;
    const float coordLoss  = b / msum;
    const float logitsLoss = c / msum;
    out[0] = 0.1f * coordLoss + logitsLoss;  // objective
    out[1] = coordLoss;
    out[2] = logitsLoss;
  }
}

extern "C" void kernel_launch(void* const* d_in, const int* in_sizes, int n_in,
                              void* d_out, int out_size, void* d_ws, size_t ws_size,
                              hipStream_t stream) {
  const float*   coord  = (const float*)d_in[0];
  const float*   logits = (const float*)d_in[1];
  const float*   disp   = (const float*)d_in[2];
  const uint8_t* valid  = (const uint8_t*)d_in[3];
  float* out     = (float*)d_out;
  float* partial = (float*)d_ws;   // kGrid*3 floats; fully written each call

  disp_loss_main<<<kGrid, kBlock, 0, stream>>>(coord, logits, disp, valid, partial);
  disp_loss_finalize<<<1, kBlock, 0, stream>>>(partial, out);

  (void)in_sizes; (void)n_in; (void)out_size; (void)ws_size;
}